// Boundary_Deep_Supervised_Loss_90769838833673
// MI455X (gfx1250) — compile-verified
//
#include <hip/hip_runtime.h>
#include <hip/hip_bf16.h>
#include <math.h>
#include <stdint.h>

// ---------------------------------------------------------------------------
// CDNA5 (gfx1250, wave32) boundary deep-supervised loss.
//  * V_WMMA_F32_16X16X4_F32 as the cross-lane reduction engine
//  * TENSOR_LOAD_TO_LDS (TDM) stages the EDT column tiles, with TDM padding
//    chosen so LDS column stride == H+2 dwords (== 2 mod 64 -> bank-free)
// ---------------------------------------------------------------------------

typedef float v2f __attribute__((ext_vector_type(2)));
typedef float v8f __attribute__((ext_vector_type(8)));
typedef unsigned int v4u __attribute__((ext_vector_type(4)));
typedef int v4i __attribute__((ext_vector_type(4)));
typedef int v8i __attribute__((ext_vector_type(8)));

#define BIGD 1.0e6f

// Wave-level sum of (a0 + a1) across all 32 lanes via one V_WMMA_F32_16X16X4_F32.
// A = per-lane partials as 16x4 tile, B = ones => D rows are row-sums of A;
// total = sum_m D[m,n] = s(lane) + s(lane^16).
__device__ __forceinline__ float wave_reduce_sum_wmma(float a0, float a1) {
  v2f a;    a[0] = a0;   a[1] = a1;
  v2f ones; ones[0] = 1.0f; ones[1] = 1.0f;
  v8f c = {};
  c = __builtin_amdgcn_wmma_f32_16x16x4_f32(false, a, false, ones,
                                            (short)0, c, false, false);
  float s = c[0] + c[1] + c[2] + c[3] + c[4] + c[5] + c[6] + c[7];
  s += __shfl_xor(s, 16, 32);
  return s;
}

// focal element, alpha = 1, gamma = 2, BCE log-clamp at -100
__device__ __forceinline__ float focal_elem(float p, float t) {
  float lp  = fmaxf(__logf(p), -100.0f);
  float l1p = fmaxf(__logf(1.0f - p), -100.0f);
  float ce  = -(t * lp + (1.0f - t) * l1p);
  float pt  = p * t + (1.0f - p) * (1.0f - t);
  float om  = 1.0f - pt;
  return t * ce * om * om;   // alpha_t == t for alpha = 1
}

__global__ void init_kernel(float* out, int* flags, int nf) {
  int i = blockIdx.x * blockDim.x + threadIdx.x;
  if (i == 0) out[0] = 0.0f;
  if (i < nf) flags[i] = 0;
}

// ---------------------------------------------------------------------------
// Focal loss over all B*2*h*w elements (sum reduction), weight wscale.
// ---------------------------------------------------------------------------
__global__ void focal_kernel(const float* __restrict__ logits,
                             const float* __restrict__ target,
                             float* __restrict__ out,
                             int h, int w, int s, float wscale) {
  const int  hw = h * w;
  const long n  = 16L * hw;
  const long stride = (long)gridDim.x * blockDim.x;
  float acc0 = 0.0f, acc1 = 0.0f;
  for (long i = (long)blockIdx.x * blockDim.x + threadIdx.x; i < n; i += stride) {
    int b = (int)(i / hw);
    int r = (int)(i - (long)b * hw);
    int y = r / w;
    int x = r - y * w;
    long lbase = ((long)(b * 2) * h + y) * w + x;
    float x0 = logits[lbase];
    float x1 = logits[lbase + (long)hw];
    long tbase = ((long)(b * 2) * 512 + ((long)y << s)) * 512 + ((long)x << s);
    float t0 = target[tbase];
    float t1 = target[tbase + 512L * 512L];
    float p1 = 1.0f / (1.0f + __expf(x0 - x1));
    float p0 = 1.0f - p1;
    acc0 += focal_elem(p0, t0);
    acc1 += focal_elem(p1, t1);
  }
  float tot = wave_reduce_sum_wmma(acc0, acc1);
  if ((threadIdx.x & 31) == 0) atomicAdd(out, wscale * tot);
}

// ---------------------------------------------------------------------------
// Row pass of the exact EDT for BOTH masks at once.  Output stored TRANSPOSED
// f2[((b*w + x)*h) + y] so stores here and TDM tile reads are contiguous.
// ---------------------------------------------------------------------------
__global__ void edt_row_kernel(const float* __restrict__ target,
                               float* __restrict__ f2p,
                               float* __restrict__ f2n,
                               int* __restrict__ flags,
                               int h, int w, int s) {
  int j = blockIdx.x * blockDim.x + threadIdx.x;
  if (j >= 16 * h) return;
  int b = j / h;
  int y = j - b * h;
  const float* trow = target + ((long)(b * 2 + 1) * 512 + ((long)y << s)) * 512;
  long idx0 = (long)b * w * h + y;

  float dp = BIGD, dn = BIGD;
  int any = 0;
  long idx = idx0;
  for (int x = 0; x < w; ++x, idx += h) {
    if (x + 16 < w) __builtin_prefetch(trow + ((long)(x + 16) << s), 0, 1);
    bool pos = trow[(long)x << s] > 0.5f;
    dp = pos ? 0.0f : dp + 1.0f;      // feature = posmask
    dn = pos ? dn + 1.0f : 0.0f;      // feature = negmask
    f2p[idx] = dp;
    f2n[idx] = dn;
    any |= (int)pos;
  }
  if (any) atomicOr(&flags[b], 1);

  dp = BIGD; dn = BIGD;
  idx = idx0 + (long)(w - 1) * h;
  for (int x = w - 1; x >= 0; --x, idx -= h) {
    bool pos = trow[(long)x << s] > 0.5f;
    dp = pos ? 0.0f : dp + 1.0f;
    dn = pos ? dn + 1.0f : 0.0f;
    float mp = fminf(f2p[idx], dp);
    float mn = fminf(f2n[idx], dn);
    f2p[idx] = mp * mp;
    f2n[idx] = mn * mn;
  }
}

// ---------------------------------------------------------------------------
// Column pass (exact min over parabolas) fused with SDF + boundary term.
// Block = (8,32); tile = 8 columns x H rows staged by the Tensor Data Mover.
// TDM pad: interval = H/2 dwords, amount = 1 dword -> LDS column stride H+2.
// ---------------------------------------------------------------------------
template <int H>
__global__ void edt_col_boundary_kernel(const float* __restrict__ logits,
                                        const float* __restrict__ target,
                                        const float* __restrict__ f2p,
                                        const float* __restrict__ f2n,
                                        const int* __restrict__ flags,
                                        float* __restrict__ out,
                                        int w, int s, float cscale) {
  constexpr int TW  = 8;
  constexpr int PST = H + 2;                                 // padded col stride
  constexpr int IVL = (H == 512) ? 7 : (H == 256) ? 6 : 5;   // 2<<IVL = H/2 dwords
  constexpr int N   = TW * H;                                // elements per tile
  __shared__ float sPl[TW * PST];
  __shared__ float sNl[TW * PST];

  const int b  = blockIdx.y;
  const int x0 = blockIdx.x * TW;
  const int tx = threadIdx.x;          // 0..7  (column)
  const int ty = threadIdx.y;          // 0..31
  const int tid = ty * TW + tx;

  const float* colP = f2p + ((long)b * w + x0) * H;  // 8 contiguous columns
  const float* colN = f2n + ((long)b * w + x0) * H;

  if (tid < 32) {                       // one wave issues both TDM descriptors
    v8i g1;
    g1[0] = (int)((2u << 16)            // data_size = 4 bytes
                  | (1u << 20)          // pad_enable
                  | ((unsigned)IVL << 22));   // pad_interval, pad_amount=1 dword
    g1[1] = (int)(((unsigned)N & 0xFFFFu) << 16);  // tensor_dim0 lo16
    g1[2] = (int)(1u << 16);                        // tensor_dim1 = 1
    g1[3] = (int)(((unsigned)N & 0xFFFFu) << 16);  // tile_dim0 = N
    g1[4] = 1;                                      // tile_dim1 = 1
    g1[5] = N;                                      // tensor_dim0_stride lo32
    g1[6] = 0;
    g1[7] = 0;
    unsigned long long gp = (unsigned long long)(uintptr_t)colP;
    unsigned long long gn = (unsigned long long)(uintptr_t)colN;
    v4u g0p, g0n;
    g0p[0] = 1u;                                         // count = 1
    g0p[1] = (unsigned)(uintptr_t)&sPl[0];               // LDS byte offset
    g0p[2] = (unsigned)(gp & 0xFFFFFFFFull);
    g0p[3] = (unsigned)((gp >> 32) & 0x1FFFFFFull) | (2u << 30);  // type=2
    g0n = g0p;
    g0n[1] = (unsigned)(uintptr_t)&sNl[0];
    g0n[2] = (unsigned)(gn & 0xFFFFFFFFull);
    g0n[3] = (unsigned)((gn >> 32) & 0x1FFFFFFull) | (2u << 30);
    v4i z4 = {};
#if defined(__clang_major__) && (__clang_major__ >= 23)
    v8i z8 = {};
    __builtin_amdgcn_tensor_load_to_lds(g0p, g1, z4, z4, z8, 0);
    __builtin_amdgcn_tensor_load_to_lds(g0n, g1, z4, z4, z8, 0);
#else
    __builtin_amdgcn_tensor_load_to_lds(g0p, g1, z4, z4, 0);
    __builtin_amdgcn_tensor_load_to_lds(g0n, g1, z4, z4, 0);
#endif
    __builtin_amdgcn_s_wait_tensorcnt(0);
  }
  __syncthreads();

  const float fg = (flags[b] != 0) ? 1.0f : 0.0f;
  const int x = x0 + tx;
  const float* pcol = sPl + tx * PST;   // padded column base
  const float* ncol = sNl + tx * PST;
  float acc0 = 0.0f, acc1 = 0.0f;

  for (int y = ty; y < H; y += 32) {
    float mp = 3.0e38f, mn = 3.0e38f;
#pragma unroll 4
    for (int yp = 0; yp < H / 2; ++yp) {        // first half: no pad offset
      float dy = (float)(y - yp);
      mp = fminf(mp, fmaf(dy, dy, pcol[yp]));
      mn = fminf(mn, fmaf(dy, dy, ncol[yp]));
    }
#pragma unroll 4
    for (int yp = H / 2; yp < H; ++yp) {        // second half: +1 pad dword
      float dy = (float)(y - yp);
      mp = fminf(mp, fmaf(dy, dy, pcol[yp + 1]));
      mn = fminf(mn, fmaf(dy, dy, ncol[yp + 1]));
    }
    float negdis = sqrtf(mp);   // EDT(posmask)
    float posdis = sqrtf(mn);   // EDT(negmask)
    bool pos = target[((long)(b * 2 + 1) * 512 + ((long)y << s)) * 512 +
                      ((long)x << s)] > 0.5f;
    float sdf = pos ? (1.0f - posdis) : negdis;
    sdf *= fg;
    long lbase = ((long)(b * 2) * H + y) * w + x;
    float xl0 = logits[lbase];
    float xl1 = logits[lbase + (long)H * w];
    float pc  = 1.0f / (1.0f + __expf(xl0 - xl1));
    float v = pc * sdf;
    if (y & 32) acc1 += v; else acc0 += v;
  }

  float tot = wave_reduce_sum_wmma(acc0, acc1);
  if ((tid & 31) == 0) atomicAdd(out, cscale * tot);
}

// ---------------------------------------------------------------------------
// Host-side orchestration (all launches on `stream`, graph-capture safe).
// ---------------------------------------------------------------------------
static void run_scale(const float* logits, const float* target, float* out,
                      float* f2p, float* f2n, int* flags, int s,
                      hipStream_t stream) {
  const int h = 512 >> s, w = 512 >> s;
  const float wsc = 1.0f / (float)(1 << s);
  const long n = 16L * h * w;

  long fb = (n + 255) / 256; if (fb > 4096) fb = 4096;
  focal_kernel<<<dim3((unsigned)fb), dim3(256), 0, stream>>>(
      logits, target, out, h, w, s, wsc);

  int rows = 16 * h;
  edt_row_kernel<<<dim3((rows + 255) / 256), dim3(256), 0, stream>>>(
      target, f2p, f2n, flags, h, w, s);

  dim3 grid(w / 8, 16), blk(8, 32);
  float csc = 0.01f * wsc / (float)n;
  switch (s) {
    case 0:
      edt_col_boundary_kernel<512><<<grid, blk, 0, stream>>>(
          logits, target, f2p, f2n, flags, out, w, s, csc);
      break;
    case 1:
      edt_col_boundary_kernel<256><<<grid, blk, 0, stream>>>(
          logits, target, f2p, f2n, flags, out, w, s, csc);
      break;
    default:
      edt_col_boundary_kernel<128><<<grid, blk, 0, stream>>>(
          logits, target, f2p, f2n, flags, out, w, s, csc);
      break;
  }
}

extern "C" void kernel_launch(void* const* d_in, const int* in_sizes, int n_in,
                              void* d_out, int out_size, void* d_ws, size_t ws_size,
                              hipStream_t stream) {
  const float* out0   = (const float*)d_in[0];   // [16,2,512,512]
  const float* out1   = (const float*)d_in[1];   // [16,2,256,256]
  const float* out2   = (const float*)d_in[2];   // [16,2,128,128]
  const float* target = (const float*)d_in[3];   // [16,2,512,512]
  float* out = (float*)d_out;

  const size_t planeMax = (size_t)16 * 512 * 512;
  float* f2p   = (float*)d_ws;
  float* f2n   = f2p + planeMax;
  int*   flags = (int*)(f2n + planeMax);         // 48 ints (16 per scale)

  init_kernel<<<dim3(1), dim3(64), 0, stream>>>(out, flags, 48);

  run_scale(out0, target, out, f2p, f2n, flags +  0, 0, stream);
  run_scale(out1, target, out, f2p, f2n, flags + 16, 1, stream);
  run_scale(out2, target, out, f2p, f2n, flags + 32, 2, stream);
}